// FactorizedSpectralConv2d_34660386078727
// MI455X (gfx1250) — compile-verified
//
#include <hip/hip_runtime.h>
#include <math.h>

// ---------------------------------------------------------------------------
// FactorizedSpectralConv2d on gfx1250 (MI455X), f32 WMMA pipeline.
//   Branch Y (width):  Xf = x · F       (GEMM, K=128)
//                      Yf = mix(Xf,w0)  (per-mode complex GEMM, K=64)
//                      out = Yf · G     (GEMM, K=64)
//   Branch X (height): same kernels with transposed (templated) strides.
// All intra-batch strides are compile-time constants so global loads use
// immediate offsets and merge into b64 where contiguous.
// ---------------------------------------------------------------------------

typedef __attribute__((ext_vector_type(2))) float v2f;
typedef __attribute__((ext_vector_type(8))) float v8f;

static __device__ __forceinline__ v8f wmma_f32(v2f a, v2f b, v8f c) {
  return __builtin_amdgcn_wmma_f32_16x16x4_f32(false, a, false, b,
                                               (short)0, c, false, false);
}

// Build the shared forward/inverse truncated-DFT bases in workspace.
__global__ void fsc_init_basis(float* __restrict__ F, float* __restrict__ G) {
  const float inv = 0.08838834764831845f;          // 1/sqrt(128)
  const float w   = 6.283185307179586f / 128.0f;   // 2*pi/128
  const int total = 128 * 64 + 64 * 128;
  for (int t = threadIdx.x + blockIdx.x * blockDim.x; t < total;
       t += blockDim.x * gridDim.x) {
    if (t < 128 * 64) {                 // F[n][j]
      int n = t >> 6, j = t & 63;
      float v = (j < 32) ? cosf(w * (float)(j * n)) * inv
                         : -sinf(w * (float)((j - 32) * n)) * inv;
      F[n * 64 + j] = v;
    } else {                            // G[j][n]
      int t2 = t - 128 * 64;
      int j = t2 >> 7, n = t2 & 127;
      float v;
      if (j < 32) {
        float a = (j == 0) ? 1.0f : 2.0f;
        v = a * cosf(w * (float)(j * n)) * inv;
      } else {
        int k = j - 32;                 // sin row; k==0 row is all zeros
        v = -2.0f * sinf(w * (float)(k * n)) * inv;
      }
      G[j * 128 + n] = v;
    }
  }
}

// Generic batched GEMM with compile-time strides: C[g] (+)= A[g] * B[g].
// Wave -> 16x16 tile via v_wmma_f32_16x16x4_f32; workgroup (8 waves) -> 32x64.
template <int K, int sAm, int sAk, int sBk, int sBn, int sCm, int sCn, bool ADD>
__global__ void __launch_bounds__(256)
fsc_wmma_gemm(const float* __restrict__ A, const float* __restrict__ B,
              float* __restrict__ C, long sAb, long sBb, long sCb) {
  const int lane = threadIdx.x & 31;
  const int wave = threadIdx.x >> 5;
  const int half = lane >> 4;          // 0: lanes 0-15, 1: lanes 16-31
  const int lr   = lane & 15;
  const int m0 = blockIdx.x * 32 + (wave & 1) * 16;
  const int n0 = blockIdx.y * 64 + (wave >> 1) * 16;
  const long g = blockIdx.z;

  // Per-lane base pointers; all K-step offsets below are compile-time imms.
  const float* Ap = A + g * sAb + (long)(m0 + lr) * sAm + (long)(2 * half) * sAk;
  const float* Bp = B + g * sBb + (long)(n0 + lr) * sBn + (long)(2 * half) * sBk;

  v8f acc = {0.f, 0.f, 0.f, 0.f, 0.f, 0.f, 0.f, 0.f};

#pragma unroll
  for (int k0 = 0; k0 < K; k0 += 4) {
    v2f a, b;
    if (sAk == 1) {
      // contiguous pair, 8B aligned (k-base even, row strides multiple of 64)
      a = *(const v2f*)(Ap + k0);
    } else {
      a.x = Ap[(long)k0 * sAk];
      a.y = Ap[(long)(k0 + 1) * sAk];
    }
    b.x = Bp[(long)k0 * sBk];
    b.y = Bp[(long)(k0 + 1) * sBk];
    acc = wmma_f32(a, b, acc);
  }

  float* Cp = C + g * sCb + (long)(m0 + 8 * half) * sCm + (long)(n0 + lr) * sCn;
#pragma unroll
  for (int j = 0; j < 8; ++j) {
    float v = acc[j];
    if (ADD) v += Cp[(long)j * sCm];
    Cp[(long)j * sCm] = v;
  }
}

// Per-mode complex channel mix: for batch b (grid.y/32), mode k (grid.y%32):
//   Yr = Xr*Wr - Xi*Wi ; Yi = Xr*Wi + Xi*Wr
// M = 128 spatial, N = 64 out-channels, K = 64 in-channels.
// Weights W[i][o][k][2] interleaved re/im (strides 4096 / 64 / 2).
// Spectral tensors share layout for X and Y: batch stride 524288,
// mode stride sK2, spatial stride sM, channel stride 8192, imag offset IMG.
template <int sK2, int sM, int IMG>
__global__ void __launch_bounds__(256)
fsc_wmma_cmix(const float* __restrict__ X, const float* __restrict__ W,
              float* __restrict__ Y) {
  const int lane = threadIdx.x & 31;
  const int wave = threadIdx.x >> 5;
  const int half = lane >> 4;
  const int lr   = lane & 15;
  const int m0 = blockIdx.x * 32 + (wave & 1) * 16;
  const int n0 = (wave >> 1) * 16;
  const int b  = blockIdx.y >> 5;      // batch
  const int km = blockIdx.y & 31;      // mode

  const float* Ar = X + (long)b * 524288 + (long)km * sK2
                      + (long)(m0 + lr) * sM + (long)(2 * half) * 8192;
  const float* Ai = Ar + IMG;
  const float* Br = W + (long)km * 2 + (long)(n0 + lr) * 64
                      + (long)(2 * half) * 4096;
  const float* Bi = Br + 1;

  v8f accR = {0.f, 0.f, 0.f, 0.f, 0.f, 0.f, 0.f, 0.f};
  v8f accI = {0.f, 0.f, 0.f, 0.f, 0.f, 0.f, 0.f, 0.f};

#pragma unroll
  for (int i0 = 0; i0 < 64; i0 += 4) {
    v2f ar, ai, br, bi, nbi;
    ar.x = Ar[(long)i0 * 8192];
    ar.y = Ar[(long)(i0 + 1) * 8192];
    ai.x = Ai[(long)i0 * 8192];
    ai.y = Ai[(long)(i0 + 1) * 8192];
    br.x = Br[(long)i0 * 4096];
    br.y = Br[(long)(i0 + 1) * 4096];
    bi.x = Bi[(long)i0 * 4096];
    bi.y = Bi[(long)(i0 + 1) * 4096];
    nbi.x = -bi.x;  nbi.y = -bi.y;     // f32 WMMA has no A/B negate
    accR = wmma_f32(ar, br,  accR);
    accR = wmma_f32(ai, nbi, accR);
    accI = wmma_f32(ar, bi,  accI);
    accI = wmma_f32(ai, br,  accI);
  }

  float* Yr = Y + (long)b * 524288 + (long)km * sK2
                + (long)(m0 + 8 * half) * sM + (long)(n0 + lr) * 8192;
  float* Yi = Yr + IMG;
#pragma unroll
  for (int j = 0; j < 8; ++j) {
    Yr[(long)j * sM] = accR[j];
    Yi[(long)j * sM] = accI[j];
  }
}

extern "C" void kernel_launch(void* const* d_in, const int* in_sizes, int n_in,
                              void* d_out, int out_size, void* d_ws, size_t ws_size,
                              hipStream_t stream) {
  const float* x  = (const float*)d_in[0];   // [32,64,128,128]
  const float* w0 = (const float*)d_in[1];   // [64,64,32,2]
  const float* w1 = (const float*)d_in[2];   // [64,64,32,2]
  float* out = (float*)d_out;                // [32,64,128,128]
  float* ws  = (float*)d_ws;

  float* F  = ws;                 //  8192 floats: forward DFT basis [128][64]
  float* G  = ws + 8192;          //  8192 floats: inverse DFT basis [64][128]
  float* Xf = ws + 16384;         //  16M floats: spectral input  (64 MB)
  float* Yf = Xf + 16777216L;     //  16M floats: spectral mixed  (64 MB)

  fsc_init_basis<<<32, 256, 0, stream>>>(F, G);

  // ---------------- Branch Y (transform along last axis, width) ------------
  // Y1: Xf[(b,i,x), j] = sum_n x[(b,i,x), n] * F[n, j]   M=262144 N=64 K=128
  fsc_wmma_gemm<128, 128, 1, 64, 1, 64, 1, false>
      <<<dim3(262144 / 32, 1, 1), 256, 0, stream>>>(x, F, Xf, 0L, 0L, 0L);
  // Y2: per (b,k): complex mix over channels, Xf -> Yf
  fsc_wmma_cmix<1, 64, 32>
      <<<dim3(4, 1024), 256, 0, stream>>>(Xf, w0, Yf);
  // Y3: out[(b,o,x), n] = sum_j Yf[(b,o,x), j] * G[j, n]   M=262144 N=128 K=64
  fsc_wmma_gemm<64, 64, 1, 128, 1, 128, 1, false>
      <<<dim3(262144 / 32, 2, 1), 256, 0, stream>>>(Yf, G, out, 0L, 0L, 0L);

  // ---------------- Branch X (transform along axis -2, height) -------------
  // X1: per (b,i): Xf[j, n] = sum_m F[m, j] * x[m, n]   M=64 N=128 K=128 nb=2048
  fsc_wmma_gemm<128, 1, 64, 128, 1, 128, 1, false>
      <<<dim3(2, 2, 2048), 256, 0, stream>>>(F, x, Xf, 0L, 16384L, 8192L);
  // X2: per (b,k): complex mix over channels (rows = n, contiguous)
  fsc_wmma_cmix<128, 1, 4096>
      <<<dim3(4, 1024), 256, 0, stream>>>(Xf, w1, Yf);
  // X3: per (b,o): out[m, n] += sum_j G[j, m] * Yf[j, n]   M=128 N=128 K=64 nb=2048
  fsc_wmma_gemm<64, 1, 128, 128, 1, 128, 1, true>
      <<<dim3(4, 2, 2048), 256, 0, stream>>>(G, Yf, out, 0L, 8192L, 16384L);
}